// HypersphericalGraphVAE_55138790146371
// MI455X (gfx1250) — compile-verified
//
#include <hip/hip_runtime.h>
#include <hip/hip_bf16.h>
#include <math.h>

#define N_NODES  50000
#define N_EDGES  800000
#define EP_EDGES 400000
#define IN_DIM   128
#define HIDDEN   256
#define LATENT   64
#define EPS      1e-8f

typedef __attribute__((ext_vector_type(2))) float v2f;
typedef __attribute__((ext_vector_type(8))) float v8f;

// ---------------------------------------------------------------- CSR build: degree count
__global__ void degi_kernel(const int* __restrict__ dstv, int* __restrict__ deg, int E) {
    int e = blockIdx.x * blockDim.x + threadIdx.x;
    if (e < E) atomicAdd(&deg[dstv[e]], 1);
}

// dinv[i] = rsqrt(deg_in(i) + 1)   (self loop; deg>=1 so reference max() is a no-op)
__global__ void dinv_kernel(const int* __restrict__ deg, float* __restrict__ dinv, int n) {
    int i = blockIdx.x * blockDim.x + threadIdx.x;
    if (i < n) dinv[i] = rsqrtf((float)deg[i] + 1.0f);
}

// Exclusive scan of deg[0..n) -> offs[0..n). Single block, 1024 threads, chunked.
__global__ void scan_kernel(const int* __restrict__ deg, int* __restrict__ offs, int n) {
    __shared__ int part[1024];
    int t = threadIdx.x;
    const int CH = (n + 1023) / 1024;
    int begin = t * CH;
    int end   = begin + CH; if (end > n) end = n; if (begin > n) begin = n;
    int s = 0;
    for (int i = begin; i < end; ++i) s += deg[i];
    part[t] = s;
    __syncthreads();
    for (int off = 1; off < 1024; off <<= 1) {
        int v = (t >= off) ? part[t - off] : 0;
        __syncthreads();
        part[t] += v;
        __syncthreads();
    }
    int excl = (t == 0) ? 0 : part[t - 1];
    for (int i = begin; i < end; ++i) {
        offs[i] = excl;
        excl += deg[i];
    }
}

// Fill CSR: for each edge, slot under its destination; store src index and dinv[src].
__global__ void fill_kernel(const int* __restrict__ srcv, const int* __restrict__ dstv,
                            const int* __restrict__ offs, int* __restrict__ cursor,
                            const float* __restrict__ dinv,
                            int* __restrict__ csr_src, float* __restrict__ csr_w, int E) {
    int e = blockIdx.x * blockDim.x + threadIdx.x;
    if (e >= E) return;
    int s = srcv[e], d = dstv[e];
    int p = offs[d] + atomicAdd(&cursor[d], 1);
    csr_src[p] = s;
    csr_w[p]   = dinv[s];
}

// ---------------------------------------------------------------- WMMA GEMM (fp32, 16x16x4)
// C[M x Ncols] = A[M x K] @ B[K x Ncols], row-major. One wave per 16x16 tile of C.
__global__ void wmma_gemm_f32(const float* __restrict__ A, const float* __restrict__ B,
                              float* __restrict__ C, int M, int K, int Ncols) {
    int wv      = (blockIdx.x * blockDim.x + threadIdx.x) >> 5;
    int tiles_n = Ncols >> 4;
    int tiles   = (M >> 4) * tiles_n;
    if (wv >= tiles) return;                 // wave-uniform: EXEC stays all-1s for WMMA
    int tm   = wv / tiles_n;
    int tn   = wv - tm * tiles_n;
    int lane = threadIdx.x & 31;
    int half = lane >> 4;                    // 0: K={0,1}; 1: K={2,3} within each 4-step
    int lo   = lane & 15;

    const float* Arow = A + (size_t)(tm * 16 + lo) * K;   // A: lane carries M
    int bcol = tn * 16 + lo;                               // B: lane carries N

    v8f acc = {};
    for (int k = 0; k < K; k += 4) {
        int kb = k + half * 2;
        float2 av = *(const float2*)(Arow + kb);           // 8B-aligned (kb even)
        v2f a; a.x = av.x; a.y = av.y;
        v2f b;
        b.x = B[(size_t)kb * Ncols + bcol];
        b.y = B[(size_t)(kb + 1) * Ncols + bcol];
        acc = __builtin_amdgcn_wmma_f32_16x16x4_f32(false, a, false, b, (short)0, acc, false, false);
    }
    float* Cp = C + (size_t)(tm * 16 + half * 8) * Ncols + tn * 16 + lo;
#pragma unroll
    for (int r = 0; r < 8; ++r) Cp[(size_t)r * Ncols] = acc[r];
}

// ---------------------------------------------------------------- conv1 gather (CSR) + self-loop + bias + relu
// One wave per node; lane holds 8 features (2 x float4). No atomics.
__global__ void gather256(const int* __restrict__ offs, const int* __restrict__ deg,
                          const int* __restrict__ csr_src, const float* __restrict__ csr_w,
                          const float* __restrict__ dinv, const float* __restrict__ h,
                          const float* __restrict__ bias, float* __restrict__ g, int n) {
    int wv = (blockIdx.x * blockDim.x + threadIdx.x) >> 5;
    if (wv >= n) return;
    int lane = threadIdx.x & 31;
    float dd = dinv[wv];

    const float4* hrow = (const float4*)(h + (size_t)wv * HIDDEN);
    float4 a0 = hrow[lane], a1 = hrow[lane + 32];
    float sw = dd * dd;                       // self-loop weight
    a0.x *= sw; a0.y *= sw; a0.z *= sw; a0.w *= sw;
    a1.x *= sw; a1.y *= sw; a1.z *= sw; a1.w *= sw;

    int beg = offs[wv], cnt = deg[wv];
    for (int j = beg; j < beg + cnt; ++j) {
        int   s = csr_src[j];
        float w = dd * csr_w[j];
        const float4* hs = (const float4*)(h + (size_t)s * HIDDEN);
        float4 v0 = hs[lane], v1 = hs[lane + 32];
        a0.x += w * v0.x; a0.y += w * v0.y; a0.z += w * v0.z; a0.w += w * v0.w;
        a1.x += w * v1.x; a1.y += w * v1.y; a1.z += w * v1.z; a1.w += w * v1.w;
    }
    const float4* bv = (const float4*)bias;
    float4 b0 = bv[lane], b1 = bv[lane + 32];
    float4 o0, o1;
    o0.x = fmaxf(a0.x + b0.x, 0.f); o0.y = fmaxf(a0.y + b0.y, 0.f);
    o0.z = fmaxf(a0.z + b0.z, 0.f); o0.w = fmaxf(a0.w + b0.w, 0.f);
    o1.x = fmaxf(a1.x + b1.x, 0.f); o1.y = fmaxf(a1.y + b1.y, 0.f);
    o1.z = fmaxf(a1.z + b1.z, 0.f); o1.w = fmaxf(a1.w + b1.w, 0.f);
    float4* grow = (float4*)(g + (size_t)wv * HIDDEN);
    grow[lane] = o0; grow[lane + 32] = o1;
}

// ---------------------------------------------------------------- conv2 gather + bias + double sphere projection
// One wave per node; lane holds 2 features. Writes mu/z to d_out (scalar: d_out+3 only 4B-aligned)
// and z to an aligned workspace copy; accumulates per-feature z sums via LDS.
__global__ void gather_norm64(const int* __restrict__ offs, const int* __restrict__ deg,
                              const int* __restrict__ csr_src, const float* __restrict__ csr_w,
                              const float* __restrict__ dinv, const float* __restrict__ h,
                              const float* __restrict__ bias,
                              float* __restrict__ mu_out, float* __restrict__ z_out,
                              float* __restrict__ z_ws, float* __restrict__ zsum, int n) {
    __shared__ float sz[LATENT];
    int tid  = threadIdx.x;
    int wv   = (blockIdx.x * blockDim.x + tid) >> 5;
    int lane = tid & 31;
    if (tid < LATENT) sz[tid] = 0.f;
    __syncthreads();
    if (wv < n) {
        float dd = dinv[wv];
        float2 acc = ((const float2*)(h + (size_t)wv * LATENT))[lane];
        float sw = dd * dd;
        acc.x *= sw; acc.y *= sw;
        int beg = offs[wv], cnt = deg[wv];
        for (int j = beg; j < beg + cnt; ++j) {
            int   s = csr_src[j];
            float w = dd * csr_w[j];
            float2 v = ((const float2*)(h + (size_t)s * LATENT))[lane];
            acc.x += w * v.x; acc.y += w * v.y;
        }
        float2 bv = ((const float2*)bias)[lane];
        float tx = acc.x + bv.x, ty = acc.y + bv.y;
        float ss = tx * tx + ty * ty;
#pragma unroll
        for (int o = 16; o > 0; o >>= 1) ss += __shfl_xor(ss, o, 32);
        float nrm  = sqrtf(ss);
        float inv1 = 1.0f / (nrm + EPS);
        float mux = tx * inv1, muy = ty * inv1;
        float nm   = nrm * inv1;                 // == ||mu||
        float inv2 = 1.0f / (nm + EPS);
        float zx = mux * inv2, zy = muy * inv2;
        size_t base = (size_t)wv * LATENT + lane * 2;
        mu_out[base]    = mux;  mu_out[base + 1] = muy;
        z_out[base]     = zx;   z_out[base + 1]  = zy;
        ((float2*)z_ws)[base >> 1] = make_float2(zx, zy);
        atomicAdd(&sz[lane * 2 + 0], zx);
        atomicAdd(&sz[lane * 2 + 1], zy);
    }
    __syncthreads();
    if (tid < LATENT) atomicAdd(&zsum[tid], sz[tid]);
}

// ---------------------------------------------------------------- decoder + BCE partials
__global__ void decode_kernel(const int* __restrict__ pos, const int* __restrict__ neg,
                              const float* __restrict__ z, float* __restrict__ recon_parts, int ep) {
    __shared__ float bsum;
    if (threadIdx.x == 0) bsum = 0.f;
    __syncthreads();
    int wv   = (blockIdx.x * blockDim.x + threadIdx.x) >> 5;
    int lane = threadIdx.x & 31;
    if (wv < 2 * ep) {
        int s, d; float label;
        if (wv < ep) { s = pos[wv]; d = pos[ep + wv]; label = 1.f; }
        else         { int e = wv - ep; s = neg[e]; d = neg[ep + e]; label = 0.f; }
        float2 a = ((const float2*)(z + (size_t)s * LATENT))[lane];
        float2 b = ((const float2*)(z + (size_t)d * LATENT))[lane];
        float p = a.x * b.x + a.y * b.y;
#pragma unroll
        for (int o = 16; o > 0; o >>= 1) p += __shfl_xor(p, o, 32);
        if (lane == 0) {
            float l = p;
            float term = fmaxf(l, 0.f) - l * label + log1pf(expf(-fabsf(l)));
            atomicAdd(&bsum, term);
        }
    }
    __syncthreads();
    if (threadIdx.x == 0) atomicAdd(&recon_parts[blockIdx.x & 255], bsum);
}

// ---------------------------------------------------------------- final scalars
__global__ void final_kernel(const float* __restrict__ zsum, const float* __restrict__ recon_parts,
                             float* __restrict__ out, int n, int total_edges) {
    if (blockIdx.x == 0 && threadIdx.x == 0) {
        float reg = 0.f;
        for (int f = 0; f < LATENT; ++f) { float m = zsum[f] / (float)n; reg += m * m; }
        float rs = 0.f;
        for (int i = 0; i < 256; ++i) rs += recon_parts[i];
        float recon = rs / (float)total_edges;
        out[0] = recon + reg;     // LAMBDA_UNIFORM = 1.0
        out[1] = recon;
        out[2] = reg;
    }
}

extern "C" void kernel_launch(void* const* d_in, const int* in_sizes, int n_in,
                              void* d_out, int out_size, void* d_ws, size_t ws_size,
                              hipStream_t stream) {
    const float* x    = (const float*)d_in[0];   // [N,128]
    const float* W1   = (const float*)d_in[1];   // [128,256]
    const float* b1   = (const float*)d_in[2];   // [256]
    const float* W2   = (const float*)d_in[3];   // [256,64]
    const float* b2   = (const float*)d_in[4];   // [64]
    const int*   edge = (const int*)d_in[5];     // [2,E]
    const int*   pos  = (const int*)d_in[6];     // [2,EP]
    const int*   neg  = (const int*)d_in[7];     // [2,EP]
    float* out = (float*)d_out;
    float* ws  = (float*)d_ws;

    // float workspace
    float* h1    = ws;                                   // N*256
    float* g1    = h1 + (size_t)N_NODES * HIDDEN;        // N*256
    float* h2    = g1 + (size_t)N_NODES * HIDDEN;        // N*64
    float* z_ws  = h2 + (size_t)N_NODES * LATENT;        // N*64
    float* dinv  = z_ws + (size_t)N_NODES * LATENT;      // N
    float* zsum  = dinv + N_NODES;                       // 64
    float* parts = zsum + LATENT;                        // 256
    float* csr_w = parts + 256;                          // E
    // int workspace
    int* deg_i   = (int*)(csr_w + N_EDGES);              // N
    int* offs    = deg_i + N_NODES;                      // N
    int* cursor  = offs + N_NODES;                       // N
    int* csr_src = cursor + N_NODES;                     // E

    float* mu_out = out + 3;
    float* z_out  = mu_out + (size_t)N_NODES * LATENT;

    const int* srcv = edge;
    const int* dstv = edge + N_EDGES;

    hipMemsetAsync(zsum, 0, sizeof(float) * (LATENT + 256), stream);
    hipMemsetAsync(deg_i, 0, sizeof(int) * N_NODES, stream);
    hipMemsetAsync(cursor, 0, sizeof(int) * N_NODES, stream);

    degi_kernel<<<(N_EDGES + 255) / 256, 256, 0, stream>>>(dstv, deg_i, N_EDGES);
    scan_kernel<<<1, 1024, 0, stream>>>(deg_i, offs, N_NODES);
    dinv_kernel<<<(N_NODES + 255) / 256, 256, 0, stream>>>(deg_i, dinv, N_NODES);
    fill_kernel<<<(N_EDGES + 255) / 256, 256, 0, stream>>>(srcv, dstv, offs, cursor, dinv,
                                                           csr_src, csr_w, N_EDGES);

    {   // h1 = x @ W1
        int tiles = (N_NODES / 16) * (HIDDEN / 16);
        wmma_gemm_f32<<<(tiles + 7) / 8, 256, 0, stream>>>(x, W1, h1, N_NODES, IN_DIM, HIDDEN);
    }
    gather256<<<(N_NODES * 32 + 255) / 256, 256, 0, stream>>>(offs, deg_i, csr_src, csr_w,
                                                              dinv, h1, b1, g1, N_NODES);
    {   // h2 = g1 @ W2
        int tiles = (N_NODES / 16) * (LATENT / 16);
        wmma_gemm_f32<<<(tiles + 7) / 8, 256, 0, stream>>>(g1, W2, h2, N_NODES, HIDDEN, LATENT);
    }
    gather_norm64<<<(N_NODES * 32 + 255) / 256, 256, 0, stream>>>(offs, deg_i, csr_src, csr_w,
                                                                  dinv, h2, b2,
                                                                  mu_out, z_out, z_ws, zsum, N_NODES);
    decode_kernel<<<(int)(((size_t)2 * EP_EDGES * 32 + 255) / 256), 256, 0, stream>>>(pos, neg, z_ws, parts, EP_EDGES);
    final_kernel<<<1, 32, 0, stream>>>(zsum, parts, out, N_NODES, 2 * EP_EDGES);
}